// AngleAlignmentLoss_47201690583527
// MI455X (gfx1250) — compile-verified
//
#include <hip/hip_runtime.h>
#include <math.h>

typedef float v2f __attribute__((ext_vector_type(2)));
typedef float v4f __attribute__((ext_vector_type(4)));
typedef float v8f __attribute__((ext_vector_type(8)));

#define B_ 128
#define D_ 2048
#define KC 128
#define LSTR 132          // KC + 4 pad -> 132 % 64 == 4 -> conflict-free frag loads
#define N0MAX 64

__launch_bounds__(256)
__global__ void anchor_gram_kernel(const float* __restrict__ inputs,
                                   const int* __restrict__ targets,
                                   const int* __restrict__ subs,
                                   float* __restrict__ partial) {
  __shared__ float tile[B_ * LSTR];   // K-chunk of (inputs - center_i); later reused as Gram S
  __shared__ float center[D_];
  __shared__ int   idx0[N0MAX], idx1[N0MAX];
  __shared__ int   grp[B_];
  __shared__ int   counts[4];
  __shared__ float invn[B_];
  __shared__ float red[256];

  const int i    = blockIdx.x;          // anchor row
  const int tid  = threadIdx.x;
  const int lane = tid & 31;
  const int wave = tid >> 5;
  const int half = lane >> 4;           // 0: lanes 0-15, 1: lanes 16-31
  const int lr   = lane & 15;

  // ---- build group-member and pair-index lists (honors targets/subs data) ----
  if (tid == 0) {
    int ti = targets[i], si = subs[i];
    int gc = 0, c0 = 0, c1 = 0;
    for (int j = 0; j < B_; ++j) {
      int tj = targets[j], sj = subs[j];
      if (tj == ti && sj == si) grp[gc++] = j;
      if (tj != ti && sj == 0 && c0 < N0MAX) idx0[c0++] = j;
      if (tj != ti && sj == 1 && c1 < N0MAX) idx1[c1++] = j;
    }
    counts[0] = gc; counts[1] = c0;
  }
  __syncthreads();
  const int gc = counts[0];
  const float ginv = 1.0f / (float)gc;

  // ---- center_i (D_ floats), cooperative ----
  for (int k = tid; k < D_; k += 256) {
    float c = 0.0f;
    for (int r = 0; r < gc; ++r) c += inputs[(size_t)grp[r] * D_ + k];
    center[k] = c * ginv;
  }

  // Wave owns tile-blocks b = wave and wave+8 of a 4x4 grid of 32x32 blocks,
  // each block = 2x2 WMMA tiles of 16x16.
  v8f acc[2][2][2] = {};

  for (int k0 = 0; k0 < D_; k0 += KC) {
    __syncthreads();   // previous chunk fully consumed; center ready on first pass
    // ---- stage (inputs - center) chunk: 128 rows x 128 cols, float4 per thread x16 ----
#pragma unroll
    for (int it = 0; it < 16; ++it) {
      int linear = it * 256 + tid;       // 0..4095
      int row = linear >> 5;
      int col = (linear & 31) << 2;
      v4f v = *(const v4f*)(inputs + (size_t)row * D_ + k0 + col);
      v4f c = *(const v4f*)&center[k0 + col];
      v4f d = v - c;
      *(v4f*)&tile[row * LSTR + col] = d;
    }
    __syncthreads();
    // ---- WMMA: S += D_chunk * D_chunkT ----
    for (int k = 0; k < KC; k += 4) {
      const int ko = k + 2 * half;       // A 16x4 f32 layout: VGPR0={K0,K2}, VGPR1={K1,K3}
#pragma unroll
      for (int blk = 0; blk < 2; ++blk) {
        const int b  = wave + blk * 8;
        const int br = b >> 2, bc = b & 3;
        v2f a0 = *(const v2f*)&tile[(32 * br      + lr) * LSTR + ko];
        v2f a1 = *(const v2f*)&tile[(32 * br + 16 + lr) * LSTR + ko];
        v2f b0 = *(const v2f*)&tile[(32 * bc      + lr) * LSTR + ko];
        v2f b1 = *(const v2f*)&tile[(32 * bc + 16 + lr) * LSTR + ko];
        acc[blk][0][0] = __builtin_amdgcn_wmma_f32_16x16x4_f32(
            false, a0, false, b0, (short)0, acc[blk][0][0], false, false);
        acc[blk][0][1] = __builtin_amdgcn_wmma_f32_16x16x4_f32(
            false, a0, false, b1, (short)0, acc[blk][0][1], false, false);
        acc[blk][1][0] = __builtin_amdgcn_wmma_f32_16x16x4_f32(
            false, a1, false, b0, (short)0, acc[blk][1][0], false, false);
        acc[blk][1][1] = __builtin_amdgcn_wmma_f32_16x16x4_f32(
            false, a1, false, b1, (short)0, acc[blk][1][1], false, false);
      }
    }
  }

  __syncthreads();   // all WMMA reads of tile done; reuse tile as S (128x128, stride LSTR)
  // ---- spill accumulators: C/D layout VGPR v -> M = v + 8*half, N = lane%16 ----
#pragma unroll
  for (int blk = 0; blk < 2; ++blk) {
    const int b  = wave + blk * 8;
    const int br = b >> 2, bc = b & 3;
#pragma unroll
    for (int r = 0; r < 2; ++r) {
#pragma unroll
      for (int c = 0; c < 2; ++c) {
        const int rowbase = (2 * br + r) * 16 + 8 * half;
        const int col     = (2 * bc + c) * 16 + lr;
#pragma unroll
        for (int v = 0; v < 8; ++v)
          tile[(rowbase + v) * LSTR + col] = acc[blk][r][c][v];
      }
    }
  }
  __syncthreads();

  // ---- inverse norms from diagonal: ||d_j|| = sqrt(S[j][j]) ----
  if (tid < B_) {
    float s = tile[tid * LSTR + tid];
    invn[tid] = 1.0f / fmaxf(sqrtf(s), 1e-12f);
  }
  __syncthreads();

  // ---- sum_m sum_k |G[a,k] - G[b,k]|, G[j,k] = S[j,k]*invn[j]*invn[k] ----
  const int n0c = counts[1];
  const int total = n0c * B_;
  float sum = 0.0f;
  for (int e = tid; e < total; e += 256) {
    int m = e >> 7;          // / B_
    int k = e & (B_ - 1);
    int a  = idx0[m];
    int bb = idx1[m];
    float ga = tile[a * LSTR + k] * invn[a];
    float gb = tile[bb * LSTR + k] * invn[bb];
    sum += fabsf(ga - gb) * invn[k];
  }
  red[tid] = sum;
  __syncthreads();
#pragma unroll
  for (int s = 128; s > 0; s >>= 1) {
    if (tid < s) red[tid] += red[tid + s];
    __syncthreads();
  }
  if (tid == 0) partial[i] = red[0];
}

__global__ void finalize_kernel(const float* __restrict__ partial,
                                const int* __restrict__ n0p,
                                float* __restrict__ out) {
  __shared__ float red[B_];
  const int tid = threadIdx.x;
  red[tid] = partial[tid];
  __syncthreads();
#pragma unroll
  for (int s = 64; s > 0; s >>= 1) {
    if (tid < s) red[tid] += red[tid + s];
    __syncthreads();
  }
  if (tid == 0) {
    float denom = (float)B_ * (float)n0p[0] * (float)B_;
    out[0] = red[0] / denom;
  }
}

extern "C" void kernel_launch(void* const* d_in, const int* in_sizes, int n_in,
                              void* d_out, int out_size, void* d_ws, size_t ws_size,
                              hipStream_t stream) {
  const float* inputs  = (const float*)d_in[0];
  const int*   targets = (const int*)d_in[1];
  const int*   subs    = (const int*)d_in[2];
  const int*   n0      = (const int*)d_in[3];
  float*       partial = (float*)d_ws;
  (void)in_sizes; (void)n_in; (void)out_size; (void)ws_size;

  anchor_gram_kernel<<<B_, 256, 0, stream>>>(inputs, targets, subs, partial);
  finalize_kernel<<<1, B_, 0, stream>>>(partial, n0, (float*)d_out);
}